// RLSE_58033598103733
// MI455X (gfx1250) — compile-verified
//
#include <hip/hip_runtime.h>
#include <math.h>

// ---------------------------------------------------------------------------
// Blocked RLS (Sherman-Morrison -> rank-16 Woodbury blocks) for n=64.
//   per 16-row block:  G = A_blk S ; M = I + G A_blk^T ; H = M^-1 [G | r]
//                      theta += G^T w ; S -= G^T H      (fp32 WMMA 16x16x4)
// Single workgroup (serial scan), S persistent in LDS, 4 wave32s.
// Cholesky: wave0, lane-per-column with __shfl (no barriers).
// Solves:   thread-per-RHS-column, fully in registers (no barriers).
// Staging:  GLOBAL_LOAD_ASYNC_TO_LDS_B128 double buffer (ASYNCcnt) +
//           global_prefetch two blocks ahead.
// ---------------------------------------------------------------------------

typedef __attribute__((ext_vector_type(2))) float v2f;
typedef __attribute__((ext_vector_type(8))) float v8f;
typedef __attribute__((ext_vector_type(4))) int   v4i;

#define NDIM  64
#define BM    16
#define PAD_W 68          // LDS row stride (floats): 16B-aligned rows, %64=4
#define GW    68          // G/H width: cols 0..63 data, col 64 = rhs
#define GAMMA 1000.0f

#if __has_builtin(__builtin_amdgcn_global_load_async_to_lds_b128) && \
    __has_builtin(__builtin_amdgcn_s_wait_asynccnt)
#define ASYNC_STAGE 1
typedef __attribute__((address_space(1))) v4i* gldv4_t;   // global int4*
typedef __attribute__((address_space(3))) v4i* ldsv4_t;   // LDS int4*
#else
#define ASYNC_STAGE 0
#endif

__device__ __forceinline__ void stage_block(const float* __restrict__ A,
                                            const float* __restrict__ Bv,
                                            int blk,
                                            float (&Abuf)[BM][PAD_W],
                                            float (&bbuf)[BM],
                                            int tid)
{
    const float* src = A + (size_t)blk * (BM * NDIM);
#if ASYNC_STAGE
    #pragma unroll
    for (int c = 0; c < 2; ++c) {               // 256 x b128 chunks, 128 lanes
        const int chunk = tid * 2 + c;
        const int row   = chunk >> 4;           // 16 chunks per 64-float row
        const int col   = (chunk & 15) << 2;
        __builtin_amdgcn_global_load_async_to_lds_b128(
            (gldv4_t)(src + row * NDIM + col),
            (ldsv4_t)&Abuf[row][col], 0, 0);
    }
    if (tid < 4) {                              // 16 floats of b
        __builtin_amdgcn_global_load_async_to_lds_b128(
            (gldv4_t)(Bv + (size_t)blk * BM + tid * 4),
            (ldsv4_t)&bbuf[tid * 4], 0, 0);
    }
#else
    const int row = tid >> 3, col = (tid & 7) * 8;
    #pragma unroll
    for (int j = 0; j < 8; ++j)
        Abuf[row][col + j] = src[row * NDIM + col + j];
    if (tid < BM) bbuf[tid] = Bv[(size_t)blk * BM + tid];
#endif
}

__device__ __forceinline__ void stage_wait()
{
#if ASYNC_STAGE
    __builtin_amdgcn_s_wait_asynccnt(0);
#endif
}

__launch_bounds__(128, 1)
__global__ void rlse_wmma_kernel(const float* __restrict__ A,
                                 const float* __restrict__ B,
                                 float* __restrict__ out,
                                 int nblk)
{
    __shared__ __align__(16) float S[NDIM][PAD_W];     // running SPD inverse
    __shared__ __align__(16) float Ablk[2][BM][PAD_W]; // double-buffered block
    __shared__ __align__(16) float bblk[2][BM];
    __shared__ __align__(16) float G[BM][GW];          // A_blk @ S ; col64 = r
    __shared__ __align__(16) float H[BM][GW];          // M^-1 [G | r]
    __shared__ float Mm[BM][20];                       // M / Cholesky L
    __shared__ float invL[BM];
    __shared__ float theta[NDIM];

    const int tid  = threadIdx.x;
    const int lane = tid & 31;
    const int wv   = tid >> 5;            // wave id 0..3 (wave32)
    const int m    = lane & 15;           // tile row (A) == tile col (B/C/D)
    const int hi   = lane >> 4;           // half-wave select
    const int koff = hi * 2;              // K offset per ISA 32-bit A layout

    // ---- stage first block (overlaps with S init) ----
    stage_block(A, B, 0, Ablk[0], bblk[0], tid);

    for (int idx = tid; idx < NDIM * NDIM; idx += 128) {
        int r = idx >> 6, c = idx & 63;
        S[r][c] = (r == c) ? GAMMA : 0.0f;
    }
    if (tid < NDIM) theta[tid] = 0.0f;

    stage_wait();
    __syncthreads();

    for (int blk = 0; blk < nblk; ++blk) {
        const int cur = blk & 1, nxt = cur ^ 1;
        const bool haveNext = (blk + 1 < nblk);

        if (haveNext)
            stage_block(A, B, blk + 1, Ablk[nxt], bblk[nxt], tid);
        if (blk + 2 < nblk)
            __builtin_prefetch(A + (size_t)(blk + 2) * (BM * NDIM) + tid * 8, 0, 0);

        // ========== 1) G = A_blk @ S  (wave wv -> cols [16wv,16wv+16)) =====
        {
            v8f acc = {};
            #pragma unroll
            for (int k = 0; k < NDIM; k += 4) {
                v2f a, b2;
                a.x  = Ablk[cur][m][k + koff];          // A 16x4 tile
                a.y  = Ablk[cur][m][k + koff + 1];
                b2.x = S[k + koff][wv * 16 + m];        // B 4x16 tile
                b2.y = S[k + koff + 1][wv * 16 + m];
                acc = __builtin_amdgcn_wmma_f32_16x16x4_f32(
                        false, a, false, b2, (short)0, acc, false, false);
            }
            #pragma unroll
            for (int v = 0; v < 8; ++v)
                G[v + hi * 8][wv * 16 + m] = acc[v];    // C/D layout writeback
        }
        // residual r = b - A_blk @ theta  -> G[:,64]
        if (tid < BM) {
            float s = bblk[cur][tid];
            #pragma unroll 8
            for (int k = 0; k < NDIM; ++k) s -= Ablk[cur][tid][k] * theta[k];
            G[tid][NDIM] = s;
        }
        __syncthreads();                                // [A] G, r ready

        // ========== 2) M = I + G @ A_blk^T   (wave 0 only) =================
        if (wv == 0) {
            v8f acc = {};
            #pragma unroll
            for (int k = 0; k < NDIM; k += 4) {
                v2f a, b2;
                a.x  = G[m][k + koff];
                a.y  = G[m][k + koff + 1];
                b2.x = Ablk[cur][m][k + koff];          // (A^T)[kk][n], n==m
                b2.y = Ablk[cur][m][k + koff + 1];
                acc = __builtin_amdgcn_wmma_f32_16x16x4_f32(
                        false, a, false, b2, (short)0, acc, false, false);
            }
            #pragma unroll
            for (int v = 0; v < 8; ++v) {
                int r = v + hi * 8;
                Mm[r][m] = acc[v] + (r == m ? 1.0f : 0.0f);
            }

            // ====== 3) Cholesky M = L L^T : lane j owns column j, no
            //           barriers (wave-internal DS ordering + shfl bcast) ====
            float c[BM];
            float myInv = 0.0f;
            #pragma unroll
            for (int i = 0; i < BM; ++i) c[i] = Mm[i][m];
            #pragma unroll
            for (int p = 0; p < BM; ++p) {
                float dp  = __shfl(c[p], p, 32);
                float d   = sqrtf(fmaxf(dp, 1e-30f));
                float inv = 1.0f / d;
                if (lane == p) {
                    c[p]  = d;
                    myInv = inv;
                    #pragma unroll
                    for (int i = p + 1; i < BM; ++i) c[i] *= inv;
                }
                float colp[BM];
                float Ljp = 0.0f;
                #pragma unroll
                for (int i = p; i < BM; ++i) {
                    colp[i] = __shfl(c[i], p, 32);      // broadcast column p
                    if (i == lane) Ljp = colp[i];
                }
                if (lane > p && lane < BM) {
                    #pragma unroll
                    for (int i = p + 1; i < BM; ++i)
                        if (i >= lane) c[i] -= colp[i] * Ljp;
                }
            }
            if (lane < BM) {
                #pragma unroll
                for (int i = 0; i < BM; ++i) Mm[i][lane] = c[i];
                invL[lane] = myInv;
            }
        }
        __syncthreads();                                // [C] L, invL ready

        // ========== 4) H = M^-1 [G | r] : one thread per RHS column,
        //               both substitutions fully in registers ================
        if (tid <= NDIM) {                              // 65 columns
            float y[BM];
            #pragma unroll
            for (int i = 0; i < BM; ++i) y[i] = G[i][tid];
            #pragma unroll
            for (int p = 0; p < BM; ++p) {              // L y' = y
                float s = y[p];
                #pragma unroll
                for (int j = 0; j < p; ++j) s -= Mm[p][j] * y[j];
                y[p] = s * invL[p];
            }
            #pragma unroll
            for (int p = BM - 1; p >= 0; --p) {         // L^T x = y'
                float s = y[p];
                #pragma unroll
                for (int j = p + 1; j < BM; ++j) s -= Mm[j][p] * y[j];
                y[p] = s * invL[p];
            }
            #pragma unroll
            for (int i = 0; i < BM; ++i) H[i][tid] = y[i];
        }
        __syncthreads();                                // [D] H ready

        // ========== 5) theta += G^T w   (w = H[:,64]) ======================
        if (tid < NDIM) {
            float s = theta[tid];
            #pragma unroll
            for (int i = 0; i < BM; ++i) s += G[i][tid] * H[i][NDIM];
            theta[tid] = s;
        }

        // ========== 6) S -= G^T @ H  (rank-16; wave wv owns row stripe) ====
        #pragma unroll
        for (int t = 0; t < 4; ++t) {
            const int tr = wv, tc = t;
            v8f acc;
            #pragma unroll
            for (int v = 0; v < 8; ++v)                 // C = S tile
                acc[v] = S[tr * 16 + v + hi * 8][tc * 16 + m];
            #pragma unroll
            for (int k = 0; k < BM; k += 4) {
                v2f a, b2;
                a.x  = -G[k + koff][tr * 16 + m];       // A = -(G^T) tile
                a.y  = -G[k + koff + 1][tr * 16 + m];
                b2.x = H[k + koff][tc * 16 + m];        // B = H tile
                b2.y = H[k + koff + 1][tc * 16 + m];
                acc = __builtin_amdgcn_wmma_f32_16x16x4_f32(
                        false, a, false, b2, (short)0, acc, false, false);
            }
            #pragma unroll
            for (int v = 0; v < 8; ++v)
                S[tr * 16 + v + hi * 8][tc * 16 + m] = acc[v];
        }

        stage_wait();                                   // drain async staging
        __syncthreads();                                // [E] next iteration
    }

    if (tid < NDIM) out[tid] = theta[tid];
}

extern "C" void kernel_launch(void* const* d_in, const int* in_sizes, int n_in,
                              void* d_out, int out_size, void* d_ws, size_t ws_size,
                              hipStream_t stream) {
    const float* A = (const float*)d_in[0];   // [16, 4096, 64] fp32
    const float* B = (const float*)d_in[1];   // [16, 4096]     fp32
    float* out = (float*)d_out;               // [64, 1]        fp32

    const int rows = in_sizes[1];             // batch*row = 65536
    const int nblk = rows / BM;               // 4096 serial rank-16 blocks

    rlse_wmma_kernel<<<1, 128, 0, stream>>>(A, B, out, nblk);
}